// Pyramid_63634235457697
// MI455X (gfx1250) — compile-verified
//
#include <hip/hip_runtime.h>
#include <hip/hip_bf16.h>

typedef __attribute__((ext_vector_type(2))) float v2f;
typedef __attribute__((ext_vector_type(8))) float v8f;

#define B_   64
#define S_   96
#define E_   300
#define D_   512
#define H_   256
#define SEL_ 128
#define MLP_ 1024
#define NC_  3

__device__ __forceinline__ float sigmf(float x) { return 1.0f / (1.0f + __expf(-x)); }

// ---------------------------------------------------------------------------
// Kernel A: embedding gather + encoder GEMM, fp32 WMMA 16x16x4.
// Grid: (B*S)/16 = 384 workgroups of 128 threads (4 waves).
// ---------------------------------------------------------------------------
__global__ __launch_bounds__(128) void encode_kernel(
    const int* __restrict__ sentences, const float* __restrict__ embed,
    const float* __restrict__ W_enc, const float* __restrict__ b_enc,
    float* __restrict__ states)
{
  __shared__ int   toks[16];
  __shared__ float a_lds[16 * 304];   // 300 padded to 304 (keeps 8B alignment)
  const int tid  = threadIdx.x;
  const int tile = blockIdx.x;        // 16-row tile of flattened (B*S)

  if (tid < 16) toks[tid] = sentences[tile * 16 + tid];
  __syncthreads();
  for (int i = tid; i < 16 * E_; i += 128) {
    int r = i / E_, k = i - r * E_;
    a_lds[r * 304 + k] = embed[(size_t)toks[r] * E_ + k];
  }
  __syncthreads();

  const int wid = tid >> 5, lane = tid & 31;
  const int lhi = lane >> 4, llo = lane & 15;

  for (int nti = 0; nti < 8; ++nti) {
    const int n0 = (wid * 8 + nti) * 16;
    v8f acc = {};
    #pragma unroll 5
    for (int kk = 0; kk < E_ / 4; ++kk) {
      const int k0 = kk * 4;
      // A (16x4 layout): lane<16 -> M=lane,K={0,1}; lane>=16 -> K={2,3}
      v2f av = *(const v2f*)&a_lds[llo * 304 + k0 + 2 * lhi];
      // B (4x16): lane<16 -> rows K={0,1}; lane>=16 -> rows K={2,3}, N=llo
      const float* wp = W_enc + (size_t)(k0 + 2 * lhi) * D_ + n0 + llo;
      v2f bv; bv.x = wp[0]; bv.y = wp[D_];
      acc = __builtin_amdgcn_wmma_f32_16x16x4_f32(false, av, false, bv,
                                                  (short)0, acc, false, false);
    }
    #pragma unroll
    for (int r = 0; r < 8; ++r) {
      const int m = r + 8 * lhi;          // C/D layout: VGPR r = rows r / r+8
      const int col = n0 + llo;
      states[(size_t)(tile * 16 + m) * D_ + col] = acc[r] + b_enc[col];
    }
  }
}

// ---------------------------------------------------------------------------
// One-time weight repacks (run every launch; deterministic).
// Wp4[k*256+j] = {W_comp[k][j], W_comp[k][256+j], W_comp[k][512+j], W_comp[k][768+j]}
// Wg1[k*256+j] = W_comp[k][1024+j]
// => compose step: one b128 + one b32 per k instead of five b32.
// ---------------------------------------------------------------------------
__global__ __launch_bounds__(256) void repack_comp_kernel(
    const float* __restrict__ W_comp, float4* __restrict__ Wp4,
    float* __restrict__ Wg1)
{
  const int k = blockIdx.x;       // 512
  const int j = threadIdx.x;      // 256
  const float* w = W_comp + (size_t)k * (5 * H_) + j;
  Wp4[k * H_ + j] = make_float4(w[0], w[H_], w[2 * H_], w[3 * H_]);
  Wg1[k * H_ + j] = w[4 * H_];
}

// Ws1p[kb*128+j] = {W_sel1[4kb][j], W_sel1[4kb+1][j], W_sel1[4kb+2][j], W_sel1[4kb+3][j]}
__global__ __launch_bounds__(128) void repack_sel_kernel(
    const float* __restrict__ W_sel1, float4* __restrict__ Ws1p)
{
  const int kb = blockIdx.x;      // 128
  const int j  = threadIdx.x;     // 128
  const float* w = W_sel1 + (size_t)(kb * 4) * SEL_ + j;
  Ws1p[kb * SEL_ + j] = make_float4(w[0], w[SEL_], w[2 * SEL_], w[3 * SEL_]);
}

// ---------------------------------------------------------------------------
// Kernel B: the 95-step easy-first scan. One workgroup (256 thr) per batch
// element; the full 96x512 state lives in LDS (196 KB < 320 KB/WG on CDNA5).
// Incremental scoring: full 95-pair scoring once, then only the 2 pairs
// adjacent to each merge are re-evaluated; score/slot arrays shift in LDS.
// ---------------------------------------------------------------------------
__global__ __launch_bounds__(256) void scan_kernel(
    const float* __restrict__ states_g,
    const float4* __restrict__ Wp4, const float* __restrict__ Wg1,
    const float* __restrict__ b_comp,
    const float4* __restrict__ Ws1p, const float* __restrict__ b_sel1,
    const float* __restrict__ W_sel2, const float* __restrict__ b_sel2,
    float* __restrict__ h_root)
{
  __shared__ __align__(16) float st[S_ * D_];   // 196608 B, slot-addressed
  __shared__ __align__(16) float xbuf[1024];    // 2x512 selector / compose input
  __shared__ float sc[S_];                      // pair scores (95 used)
  __shared__ int   ord[S_];                     // active slot order
  __shared__ float redf[8];
  __shared__ int   redi[8];
  __shared__ int   s_idx;

  const int tid = threadIdx.x;
  const int b   = blockIdx.x;
  const float b2 = b_sel2[0];

  {
    const float4* sg4 = (const float4*)(states_g + (size_t)b * S_ * D_);
    float4* st4 = (float4*)st;
    for (int i = tid; i < S_ * D_ / 4; i += 256) st4[i] = sg4[i];
  }
  if (tid < S_) ord[tid] = tid;
  __syncthreads();

  // Score up to two pairs: p0 by threads 0-127, p1 by threads 128-255.
  // score(p) = b2 + sum_j tanh([h_l;h_r].W1[:,j] + b1[j]) * W2[j]
  auto eval_pair = [&](int p0, int p1, int npairs) {
    const int half = tid >> 7, lt = tid & 127;
    const int p = half ? p1 : p0;
    const bool valid = (p >= 0) && (p < npairs);
    if (valid) {
      const int sa = ord[p] * D_ + H_, sb = ord[p + 1] * D_ + H_;
      for (int i = lt; i < H_; i += 128) {
        xbuf[half * 512 + i]      = st[sa + i];
        xbuf[half * 512 + H_ + i] = st[sb + i];
      }
    }
    __syncthreads();
    float partial = 0.f;
    if (valid) {
      const float* x = &xbuf[half * 512];
      float s0 = b_sel1[lt], s1 = 0.f, s2 = 0.f, s3 = 0.f;
      #pragma unroll 4
      for (int kb = 0; kb < (2 * H_) / 4; ++kb) {       // 128 b128 weight loads
        const float4 w = Ws1p[kb * SEL_ + lt];
        s0 = fmaf(x[4 * kb + 0], w.x, s0);
        s1 = fmaf(x[4 * kb + 1], w.y, s1);
        s2 = fmaf(x[4 * kb + 2], w.z, s2);
        s3 = fmaf(x[4 * kb + 3], w.w, s3);
      }
      partial = tanhf((s0 + s1) + (s2 + s3)) * W_sel2[lt];
    }
    #pragma unroll
    for (int off = 16; off > 0; off >>= 1) partial += __shfl_xor(partial, off, 32);
    if ((tid & 31) == 0) redf[tid >> 5] = partial;
    __syncthreads();
    if ((lt == 0) && valid)
      sc[p] = b2 + redf[half * 4] + redf[half * 4 + 1] +
                   redf[half * 4 + 2] + redf[half * 4 + 3];
    __syncthreads();
  };

  // one-time full scoring of all 95 adjacent pairs
  for (int p0 = 0; p0 < S_ - 1; p0 += 2) eval_pair(p0, p0 + 1, S_ - 1);

  for (int t = 0; t < S_ - 1; ++t) {
    const int len = S_ - t;       // active states
    const int n   = len - 1;      // candidate pairs
    // argmax with first-index tie-break (matches jnp.argmax)
    float v = (tid < n) ? sc[tid] : -3.4e38f;
    int  ix = (tid < n) ? tid : 0x7fffffff;
    #pragma unroll
    for (int off = 16; off > 0; off >>= 1) {
      float ov = __shfl_xor(v, off, 32);
      int   oi = __shfl_xor(ix, off, 32);
      if (ov > v || (ov == v && oi < ix)) { v = ov; ix = oi; }
    }
    if ((tid & 31) == 0) { redf[tid >> 5] = v; redi[tid >> 5] = ix; }
    __syncthreads();
    if (tid == 0) {
      float bv = redf[0]; int bi = redi[0];
      for (int w = 1; w < 8; ++w)
        if (redf[w] > bv || (redf[w] == bv && redi[w] < bi)) { bv = redf[w]; bi = redi[w]; }
      s_idx = bi;
    }
    __syncthreads();
    const int idx = s_idx;
    const int sl = ord[idx], sr = ord[idx + 1];

    // stage compose input x = [h_left ; h_right] (512)
    xbuf[tid]      = st[sl * D_ + H_ + tid];
    xbuf[H_ + tid] = st[sr * D_ + H_ + tid];
    __syncthreads();

    // a = x @ W_comp + b_comp; thread j holds all 5 gates for hidden unit j
    float g0 = b_comp[tid],          g1 = b_comp[H_ + tid],
          g2 = b_comp[2 * H_ + tid], g3 = b_comp[3 * H_ + tid],
          g4 = b_comp[4 * H_ + tid];
    #pragma unroll 4
    for (int k = 0; k < 2 * H_; ++k) {
      const float xk = xbuf[k];                       // LDS broadcast
      const float4 w = Wp4[k * H_ + tid];             // b128, L2-hot
      g0 = fmaf(xk, w.x, g0);
      g1 = fmaf(xk, w.y, g1);
      g2 = fmaf(xk, w.z, g2);
      g3 = fmaf(xk, w.w, g3);
      g4 = fmaf(xk, Wg1[k * H_ + tid], g4);
    }
    const float c_l = st[sl * D_ + tid], c_r = st[sr * D_ + tid];
    const float c = sigmf(g1) * c_l + sigmf(g2) * c_r + sigmf(g0) * tanhf(g4);
    st[sl * D_ + tid]      = c;                 // composed node reuses left slot
    st[sl * D_ + H_ + tid] = sigmf(g3) * tanhf(c);

    // shift slot order / cached scores past the merge point
    int otmp = 0; float stmp = 0.f;
    const bool mo = (tid >= idx + 1) && (tid < len - 1);
    const bool ms = (tid >= idx + 1) && (tid <= len - 3);
    if (mo) otmp = ord[tid + 1];
    if (ms) stmp = sc[tid + 1];
    __syncthreads();
    if (mo) ord[tid] = otmp;
    if (ms) sc[tid] = stmp;
    __syncthreads();

    // only the two pairs touching the new node need re-scoring
    eval_pair(idx - 1, idx, n - 1);
  }

  h_root[(size_t)b * H_ + tid] = st[ord[0] * D_ + H_ + tid];
}

// ---------------------------------------------------------------------------
// Kernel C: fp32 WMMA GEMM tile kernel for the MLP head.
// One wave per 16x16 output tile; C = act(A @ W + bias).
// ---------------------------------------------------------------------------
__global__ __launch_bounds__(32) void gemm16_kernel(
    const float* __restrict__ A, int lda,
    const float* __restrict__ W, const float* __restrict__ bias,
    float* __restrict__ C, int N, int K, int do_relu)
{
  const int lane = threadIdx.x;
  const int lhi = lane >> 4, llo = lane & 15;
  const int n0 = blockIdx.x * 16;
  const int m0 = blockIdx.y * 16;
  v8f acc = {};
  for (int k0 = 0; k0 < K; k0 += 4) {
    v2f av = *(const v2f*)&A[(size_t)(m0 + llo) * lda + k0 + 2 * lhi];
    const float* wp = W + (size_t)(k0 + 2 * lhi) * N + n0 + llo;
    v2f bv; bv.x = wp[0]; bv.y = wp[N];
    acc = __builtin_amdgcn_wmma_f32_16x16x4_f32(false, av, false, bv,
                                                (short)0, acc, false, false);
  }
  #pragma unroll
  for (int r = 0; r < 8; ++r) {
    const int m = m0 + r + 8 * lhi, col = n0 + llo;
    float o = acc[r] + bias[col];
    if (do_relu) o = fmaxf(o, 0.f);
    C[(size_t)m * N + col] = o;
  }
}

// Final 64x1024 @ 1024x3 projection: tiny, VALU.
__global__ __launch_bounds__(256) void head_kernel(
    const float* __restrict__ x, const float* __restrict__ W_out,
    const float* __restrict__ b_out, float* __restrict__ out)
{
  const int t = threadIdx.x;
  if (t >= B_ * NC_) return;
  const int row = t / NC_, c = t - row * NC_;
  float acc = b_out[c];
  for (int k = 0; k < MLP_; ++k)
    acc = fmaf(x[row * MLP_ + k], W_out[k * NC_ + c], acc);
  out[t] = acc;
}

extern "C" void kernel_launch(void* const* d_in, const int* in_sizes, int n_in,
                              void* d_out, int out_size, void* d_ws, size_t ws_size,
                              hipStream_t stream) {
  const int*   sentences = (const int*)d_in[0];
  // d_in[1] = transitions: unused by the reference computation
  const float* embed  = (const float*)d_in[2];
  const float* W_enc  = (const float*)d_in[3];
  const float* b_enc  = (const float*)d_in[4];
  const float* W_comp = (const float*)d_in[5];
  const float* b_comp = (const float*)d_in[6];
  const float* W_sel1 = (const float*)d_in[7];
  const float* b_sel1 = (const float*)d_in[8];
  const float* W_sel2 = (const float*)d_in[9];
  const float* b_sel2 = (const float*)d_in[10];
  const float* W_m1   = (const float*)d_in[11];
  const float* b_m1   = (const float*)d_in[12];
  const float* W_m2   = (const float*)d_in[13];
  const float* b_m2   = (const float*)d_in[14];
  const float* W_out  = (const float*)d_in[15];
  const float* b_out  = (const float*)d_in[16];

  float* states = (float*)d_ws;                         // 64*96*512 f32 = 12.6 MB
  float* h_root = states + (size_t)B_ * S_ * D_;        // 64*256
  float* x1     = h_root + (size_t)B_ * H_;             // 64*1024
  float* x2     = x1 + (size_t)B_ * MLP_;               // 64*1024
  float* wp4    = x2 + (size_t)B_ * MLP_;               // 512*256*4 = 2 MB
  float* wg1    = wp4 + (size_t)D_ * H_ * 4;            // 512*256   = 0.5 MB
  float* ws1p   = wg1 + (size_t)D_ * H_;                // 128*128*4 = 256 KB

  encode_kernel<<<(B_ * S_) / 16, 128, 0, stream>>>(sentences, embed, W_enc, b_enc, states);
  repack_comp_kernel<<<D_, 256, 0, stream>>>(W_comp, (float4*)wp4, wg1);
  repack_sel_kernel<<<SEL_, 128, 0, stream>>>(W_sel1, (float4*)ws1p);
  scan_kernel<<<B_, 256, 0, stream>>>(states, (const float4*)wp4, wg1, b_comp,
                                      (const float4*)ws1p, b_sel1, W_sel2, b_sel2, h_root);
  gemm16_kernel<<<dim3(MLP_ / 16, B_ / 16), 32, 0, stream>>>(h_root, H_, W_m1, b_m1, x1, MLP_, H_, 1);
  gemm16_kernel<<<dim3(MLP_ / 16, B_ / 16), 32, 0, stream>>>(x1, MLP_, W_m2, b_m2, x2, MLP_, MLP_, 1);
  head_kernel<<<1, 256, 0, stream>>>(x2, W_out, b_out, (float*)d_out);
}